// Attention_12558484373675
// MI455X (gfx1250) — compile-verified
//
#include <hip/hip_runtime.h>
#include <hip/hip_bf16.h>
#include <math.h>

// Problem constants
#define S 4096
#define B 64
#define H 256
#define TWO_H 512

typedef __attribute__((ext_vector_type(2))) float v2f;
typedef __attribute__((ext_vector_type(8))) float v8f;

// ---------------------------------------------------------------------------
// fp32 WMMA: D = A(16x4) * B(4x16) + C(16x16)   (exact fp32, no precision loss)
// A layout (ISA 7.12.2): lanes 0-15 hold M=lane, VGPR0=K0/K2(hi half), VGPR1=K1/K3
// B layout: lanes 0-15 hold N=lane, VGPR0=K0 (K2 for lanes 16-31), VGPR1=K1 (K3)
// C/D layout: VGPR v holds M=v (lanes 0-15) and M=v+8 (lanes 16-31), N=lane%16
// ---------------------------------------------------------------------------
__device__ __forceinline__ v8f wmma4_f32(v2f a, v2f b, v8f c) {
    return __builtin_amdgcn_wmma_f32_16x16x4_f32(
        /*neg_a=*/false, a, /*neg_b=*/false, b,
        /*c_mod=*/(short)0, c, /*reuse_a=*/false, /*reuse_b=*/false);
}

// ---------------------------------------------------------------------------
// k0: dWa = d @ W_a  and  h = tanh(d @ W_p).  One wave per 16x16 output tile.
// blocks 0..63 -> dWa tiles; blocks 64..127 -> h tiles. M=64, N=256, K=256.
// ---------------------------------------------------------------------------
__global__ void k0_small_gemms(const float* __restrict__ d,
                               const float* __restrict__ W_a,
                               const float* __restrict__ W_p,
                               float* __restrict__ dWa,
                               float* __restrict__ hout) {
    const int  tile   = blockIdx.x & 63;
    const bool second = blockIdx.x >= 64;
    const float* W = second ? W_p : W_a;
    const int m0 = (tile & 3) * 16;
    const int n0 = (tile >> 2) * 16;
    const int lane = threadIdx.x & 31;
    const int half = lane >> 4;     // 0: K pair {0,1}, 1: K pair {2,3}
    const int l    = lane & 15;

    v8f acc = {0.f, 0.f, 0.f, 0.f, 0.f, 0.f, 0.f, 0.f};
    for (int k0 = 0; k0 < H; k0 += 4) {
        const int ka = k0 + 2 * half;
        v2f av, bv;
        av.x = d[(m0 + l) * H + ka];
        av.y = d[(m0 + l) * H + ka + 1];
        bv.x = W[(ka)     * H + n0 + l];
        bv.y = W[(ka + 1) * H + n0 + l];
        acc = wmma4_f32(av, bv, acc);
    }
    float* out = second ? hout : dWa;
    #pragma unroll
    for (int v = 0; v < 8; ++v) {
        const int m = v + 8 * half;
        float val = acc[v];
        if (second) val = tanhf(val);
        out[(m0 + m) * H + n0 + l] = val;
    }
}

// ---------------------------------------------------------------------------
// k1: the single streaming pass over e (256 MB).  scores[s,b] = dWa[b,:] . e[s,b,:]
// grid (64 s-chunks, 64 batches), 256 threads = 8 waves; one wave per e-row,
// float4 (b128) fully-coalesced loads, wave32 shuffle reduction.
// Scores written into the w-region of d_out at [s*B + b] (overwritten by k2).
// ---------------------------------------------------------------------------
__global__ void k1_scores(const float* __restrict__ e,
                          const float* __restrict__ dWa,
                          float* __restrict__ scores) {
    const int b    = blockIdx.y;
    const int wave = threadIdx.x >> 5;
    const int lane = threadIdx.x & 31;

    // each lane owns 8 consecutive h values of the query vector
    const float4* q4 = (const float4*)(dWa + b * H);
    const float4 q0 = q4[lane * 2 + 0];
    const float4 q1 = q4[lane * 2 + 1];

    const int sbase = blockIdx.x * 64 + wave * 8;
    #pragma unroll
    for (int r = 0; r < 8; ++r) {
        const int s = sbase + r;
        const float4* e4 = (const float4*)(e + ((size_t)s * B + b) * H);
        const float4 a0 = e4[lane * 2 + 0];
        const float4 a1 = e4[lane * 2 + 1];
        float p = a0.x * q0.x + a0.y * q0.y + a0.z * q0.z + a0.w * q0.w
                + a1.x * q1.x + a1.y * q1.y + a1.z * q1.z + a1.w * q1.w;
        #pragma unroll
        for (int off = 16; off > 0; off >>= 1)
            p += __shfl_xor(p, off, 32);
        if (lane == 0) scores[s * B + b] = p;
    }
}

// ---------------------------------------------------------------------------
// block reductions (256 threads)
// ---------------------------------------------------------------------------
__device__ __forceinline__ float block_sum(float v, float* red) {
    const int t = threadIdx.x;
    __syncthreads();
    red[t] = v;
    __syncthreads();
    for (int off = 128; off > 0; off >>= 1) {
        if (t < off) red[t] += red[t + off];
        __syncthreads();
    }
    return red[0];
}
__device__ __forceinline__ float block_max(float v, float* red) {
    const int t = threadIdx.x;
    __syncthreads();
    red[t] = v;
    __syncthreads();
    for (int off = 128; off > 0; off >>= 1) {
        if (t < off) red[t] = fmaxf(red[t], red[t + off]);
        __syncthreads();
    }
    return red[0];
}

// ---------------------------------------------------------------------------
// k2: one block per batch b.
//   p_center, softmax stats over 4096 scores, write w in place,
//   gather the <=5 window rows of e into context, pack x = [context, d].
// ---------------------------------------------------------------------------
__global__ void k2_softmax_window(const float* __restrict__ e,
                                  const float* __restrict__ dvec,
                                  const float* __restrict__ v_p,
                                  const float* __restrict__ h,
                                  float* __restrict__ wout,   // scores in, w out
                                  float* __restrict__ x) {    // [B, 2H]
    __shared__ float red[256];
    __shared__ float lds_w[8];
    const int b = blockIdx.x;
    const int t = threadIdx.x;

    // p_center[b] = sigmoid(h[b,:] . v_p) * S
    const float dot = block_sum(h[b * H + t] * v_p[t], red);
    const float pc  = (float)S / (1.0f + expf(-dot));

    // gather this thread's 16 scores (strided by 256 over s)
    float sc[16];
    #pragma unroll
    for (int i = 0; i < 16; ++i)
        sc[i] = wout[(t + i * 256) * B + b];

    float m = -3.4e38f;
    #pragma unroll
    for (int i = 0; i < 16; ++i) m = fmaxf(m, sc[i]);
    const float M = block_max(m, red);

    float ls = 0.f;
    #pragma unroll
    for (int i = 0; i < 16; ++i) ls += expf(sc[i] - M);
    const float SUM = block_sum(ls, red);
    const float inv = 1.0f / SUM;

    // window of nonzero p: |pc - s| <= 2  ->  at most 5 integer s
    int slo = (int)ceilf(pc - 2.0f);
    int shi = (int)floorf(pc + 2.0f);
    slo = slo < 0 ? 0 : slo;
    shi = shi > S - 1 ? S - 1 : shi;

    // write w = softmax * gaussian * mask (zeros outside window)
    #pragma unroll
    for (int i = 0; i < 16; ++i) {
        const int s = t + i * 256;
        const float a    = expf(sc[i] - M) * inv;
        const float diff = pc - (float)s;
        const float p    = (fabsf(diff) <= 2.0f) ? expf(-0.5f * diff * diff) : 0.0f;
        const float wv   = a * p;
        wout[s * B + b] = wv;
        if (s >= slo && s <= shi) lds_w[s - slo] = wv;
    }
    __syncthreads();

    // context[b, t] = sum over window rows of e[s,b,t] * w[s,b]
    float ctx = 0.f;
    for (int s = slo; s <= shi; ++s)
        ctx += e[((size_t)s * B + b) * H + t] * lds_w[s - slo];

    x[b * TWO_H + t]     = ctx;               // context half
    x[b * TWO_H + H + t] = dvec[b * H + t];   // d half
}

// ---------------------------------------------------------------------------
// k3: out = relu(x @ lin_w.T + lin_b)   M=64, N=256, K=512 — WMMA f32.
// One wave per 16x16 tile.  B(k,n) = lin_w[n*2H + k]  (lin_w is [H, 2H]).
// ---------------------------------------------------------------------------
__global__ void k3_linear(const float* __restrict__ x,
                          const float* __restrict__ lin_w,
                          const float* __restrict__ lin_b,
                          float* __restrict__ out) {
    const int tile = blockIdx.x;
    const int m0 = (tile & 3) * 16;
    const int n0 = (tile >> 2) * 16;
    const int lane = threadIdx.x & 31;
    const int half = lane >> 4;
    const int l    = lane & 15;

    v8f acc = {0.f, 0.f, 0.f, 0.f, 0.f, 0.f, 0.f, 0.f};
    for (int k0 = 0; k0 < TWO_H; k0 += 4) {
        const int ka = k0 + 2 * half;
        v2f av, bv;
        av.x = x[(m0 + l) * TWO_H + ka];
        av.y = x[(m0 + l) * TWO_H + ka + 1];
        bv.x = lin_w[(n0 + l) * TWO_H + ka];
        bv.y = lin_w[(n0 + l) * TWO_H + ka + 1];
        acc = wmma4_f32(av, bv, acc);
    }
    const float bias = lin_b[n0 + l];
    #pragma unroll
    for (int v = 0; v < 8; ++v) {
        const int m = v + 8 * half;
        out[(m0 + m) * H + n0 + l] = fmaxf(acc[v] + bias, 0.0f);
    }
}

// ---------------------------------------------------------------------------
extern "C" void kernel_launch(void* const* d_in, const int* in_sizes, int n_in,
                              void* d_out, int out_size, void* d_ws, size_t ws_size,
                              hipStream_t stream) {
    const float* e     = (const float*)d_in[0];   // [S,B,H]
    const float* d     = (const float*)d_in[1];   // [1,B,H]
    const float* W_a   = (const float*)d_in[2];   // [H,H]
    const float* W_p   = (const float*)d_in[3];   // [H,H]
    const float* v_p   = (const float*)d_in[4];   // [H,1]
    const float* lin_w = (const float*)d_in[5];   // [H,2H]
    const float* lin_b = (const float*)d_in[6];   // [H]

    float* out  = (float*)d_out;        // [1,B,H]  = 16384 floats
    float* wout = out + B * H;          // [S,B]    = 262144 floats (scores in place)

    float* ws     = (float*)d_ws;       // only 256 KB of ws used
    float* ws_dWa = ws;                 // [B,H]
    float* ws_h   = ws + B * H;         // [B,H]
    float* ws_x   = ws + 2 * B * H;     // [B,2H]

    // tiny WMMA GEMMs: dWa = d@W_a, h = tanh(d@W_p)
    k0_small_gemms<<<dim3(128), 32, 0, stream>>>(d, W_a, W_p, ws_dWa, ws_h);

    // single streaming pass over e: scores
    k1_scores<<<dim3(S / 64, B), 256, 0, stream>>>(e, ws_dWa, wout);

    // softmax + local-p window + context gather (per-batch block)
    k2_softmax_window<<<dim3(B), 256, 0, stream>>>(e, d, v_p, ws_h, wout, ws_x);

    // final WMMA GEMM + bias + relu
    k3_linear<<<dim3((B / 16) * (H / 16)), 32, 0, stream>>>(ws_x, lin_w, lin_b, out);
}